// BaseGAttN_893353198188
// MI455X (gfx1250) — compile-verified
//
#include <hip/hip_runtime.h>
#include <hip/hip_bf16.h>

// ---------------------------------------------------------------------------
// 2-layer GCN (GCNConv -> ReLU -> GCNConv -> log_softmax) for gfx1250.
// Shapes: x[N,512], W1[512,16], b1[16], W2[16,40], b2[40], edges[2,E].
// GEMMs use V_WMMA_F32_16X16X4_F32 (exact fp32 matrix core path).
// ---------------------------------------------------------------------------

#define IN_CH   512
#define HID     16
#define OUTC    40
#define W2PAD   48   // 16-col-padded width for 3 WMMA tiles

typedef __attribute__((ext_vector_type(2))) float v2f;
typedef __attribute__((ext_vector_type(8))) float v8f;

// ---------------- init: deg=1 (self loop), agg1=0, out=0 -------------------
__global__ void gcn_init_kernel(float* __restrict__ deg, float* __restrict__ agg1,
                                float* __restrict__ out, int n) {
    int total = n * OUTC;
    int stride = gridDim.x * blockDim.x;
    for (int i = blockIdx.x * blockDim.x + threadIdx.x; i < total; i += stride) {
        out[i] = 0.0f;
        if (i < n * HID) agg1[i] = 0.0f;
        if (i < n)       deg[i]  = 1.0f;   // self-loop contribution to degree
    }
}

// ---------------- degree over targets --------------------------------------
__global__ void gcn_deg_kernel(const int* __restrict__ ecol, float* __restrict__ deg, int e) {
    int stride = gridDim.x * blockDim.x;
    for (int i = blockIdx.x * blockDim.x + threadIdx.x; i < e; i += stride)
        atomicAdd(&deg[ecol[i]], 1.0f);
}

// ---------------- dis = rsqrt(deg) in place --------------------------------
__global__ void gcn_dis_kernel(float* __restrict__ dis, int n) {
    int i = blockIdx.x * blockDim.x + threadIdx.x;
    if (i < n) {
        float d = dis[i];
        dis[i] = (d > 0.0f) ? rsqrtf(d) : 0.0f;
    }
}

// ---------------- GEMM1: h1[N,16] = x[N,512] @ W1[512,16] (WMMA fp32) ------
__global__ void gcn_gemm1_kernel(const float* __restrict__ x, const float* __restrict__ W1,
                                 float* __restrict__ h1, int n_tiles) {
    __shared__ float w1s[IN_CH * HID];                     // 32 KB
    for (int i = threadIdx.x; i < IN_CH * HID; i += blockDim.x)
        w1s[i] = W1[i];
    __syncthreads();

    const int wave = threadIdx.x >> 5;
    const int lane = threadIdx.x & 31;
    const int tile = blockIdx.x * (blockDim.x >> 5) + wave;
    if (tile >= n_tiles) return;                           // wave-uniform exit

    const int hi = lane >> 4;                              // half-wave select
    const int lo = lane & 15;
    const float* xrow = x + (size_t)(tile * 16 + lo) * IN_CH;

    v8f acc = {};
    for (int k0 = 0; k0 < IN_CH; k0 += 4) {
        const int k = k0 + 2 * hi;                         // lane holds K = {k, k+1}
        v2f a = *(const v2f*)(xrow + k);                   // A: row lo, K=k..k+1
        v2f b;
        b.x = w1s[k * HID + lo];                           // B: K=k,   N=lo
        b.y = w1s[(k + 1) * HID + lo];                     // B: K=k+1, N=lo
        acc = __builtin_amdgcn_wmma_f32_16x16x4_f32(
            false, a, false, b, (short)0, acc, false, false);
    }

    const int mbase = tile * 16 + 8 * hi;                  // D VGPR g -> row mbase+g
#pragma unroll
    for (int g = 0; g < 8; ++g)
        h1[(size_t)(mbase + g) * HID + lo] = acc[g];
}

// ---------------- edge scatter layer1 (16 ch) ------------------------------
__global__ void gcn_scatter1_kernel(const int* __restrict__ erow, const int* __restrict__ ecol,
                                    const float* __restrict__ dis, const float* __restrict__ h1,
                                    float* __restrict__ agg1, int e) {
    int stride = gridDim.x * blockDim.x;
    for (int i = blockIdx.x * blockDim.x + threadIdx.x; i < e; i += stride) {
        const int r = erow[i], c = ecol[i];
        const float nrm = dis[r] * dis[c];
        const float4* hp = (const float4*)(h1 + (size_t)r * HID);
        float* ap = agg1 + (size_t)c * HID;
#pragma unroll
        for (int q = 0; q < 4; ++q) {
            float4 v = hp[q];
            atomicAdd(ap + 4 * q + 0, nrm * v.x);
            atomicAdd(ap + 4 * q + 1, nrm * v.y);
            atomicAdd(ap + 4 * q + 2, nrm * v.z);
            atomicAdd(ap + 4 * q + 3, nrm * v.w);
        }
    }
}

// ---------------- self-loop + bias + ReLU (layer1), in place on agg1 -------
__global__ void gcn_relu1_kernel(const float* __restrict__ dis, const float* __restrict__ h1,
                                 const float* __restrict__ b1, float* __restrict__ agg1, int n) {
    int i = blockIdx.x * blockDim.x + threadIdx.x;
    if (i < n * HID) {
        const int node = i >> 4;
        const int c = i & 15;
        const float sn = dis[node] * dis[node];            // self-loop norm
        const float v = agg1[i] + sn * h1[i] + b1[c];
        agg1[i] = v > 0.0f ? v : 0.0f;
    }
}

// ---------------- GEMM2: h2[N,40] = h1r[N,16] @ W2[16,40] (WMMA fp32) ------
__global__ void gcn_gemm2_kernel(const float* __restrict__ h1r, const float* __restrict__ W2,
                                 float* __restrict__ h2, int n_tiles) {
    __shared__ float w2s[HID * W2PAD];                     // 3 KB, zero padded 16x48
    for (int i = threadIdx.x; i < HID * W2PAD; i += blockDim.x) {
        const int k = i / W2PAD, c = i % W2PAD;
        w2s[i] = (c < OUTC) ? W2[k * OUTC + c] : 0.0f;
    }
    __syncthreads();

    const int wave = threadIdx.x >> 5;
    const int lane = threadIdx.x & 31;
    const int tile = blockIdx.x * (blockDim.x >> 5) + wave;
    if (tile >= n_tiles) return;                           // wave-uniform exit

    const int hi = lane >> 4;
    const int lo = lane & 15;
    const float* arow = h1r + (size_t)(tile * 16 + lo) * HID;

    for (int nt = 0; nt < 3; ++nt) {                       // 3 tiles of 16 cols (40 used)
        v8f acc = {};
#pragma unroll
        for (int k0 = 0; k0 < HID; k0 += 4) {
            const int k = k0 + 2 * hi;
            v2f a = *(const v2f*)(arow + k);
            v2f b;
            b.x = w2s[k * W2PAD + nt * 16 + lo];
            b.y = w2s[(k + 1) * W2PAD + nt * 16 + lo];
            acc = __builtin_amdgcn_wmma_f32_16x16x4_f32(
                false, a, false, b, (short)0, acc, false, false);
        }
        const int col = nt * 16 + lo;
        if (col < OUTC) {                                  // mask applied AFTER the MMAs
            const int mbase = tile * 16 + 8 * hi;
#pragma unroll
            for (int g = 0; g < 8; ++g)
                h2[(size_t)(mbase + g) * OUTC + col] = acc[g];
        }
    }
}

// ---------------- edge scatter layer2 (40 ch) into d_out -------------------
__global__ void gcn_scatter2_kernel(const int* __restrict__ erow, const int* __restrict__ ecol,
                                    const float* __restrict__ dis, const float* __restrict__ h2,
                                    float* __restrict__ out, int e) {
    int stride = gridDim.x * blockDim.x;
    for (int i = blockIdx.x * blockDim.x + threadIdx.x; i < e; i += stride) {
        const int r = erow[i], c = ecol[i];
        const float nrm = dis[r] * dis[c];
        const float4* hp = (const float4*)(h2 + (size_t)r * OUTC);
        float* op = out + (size_t)c * OUTC;
#pragma unroll
        for (int q = 0; q < 10; ++q) {
            float4 v = hp[q];
            atomicAdd(op + 4 * q + 0, nrm * v.x);
            atomicAdd(op + 4 * q + 1, nrm * v.y);
            atomicAdd(op + 4 * q + 2, nrm * v.z);
            atomicAdd(op + 4 * q + 3, nrm * v.w);
        }
    }
}

// ---------------- self-loop + bias + log_softmax, in place on out ----------
__global__ void gcn_finalize_kernel(const float* __restrict__ dis, const float* __restrict__ h2,
                                    const float* __restrict__ b2, float* __restrict__ out, int n) {
    int node = blockIdx.x * blockDim.x + threadIdx.x;
    if (node >= n) return;
    const float sn = dis[node] * dis[node];
    float v[OUTC];
    float m = -3.402823466e+38f;
#pragma unroll
    for (int c = 0; c < OUTC; ++c) {
        const float t = out[(size_t)node * OUTC + c] + sn * h2[(size_t)node * OUTC + c] + b2[c];
        v[c] = t;
        m = fmaxf(m, t);
    }
    float s = 0.0f;
#pragma unroll
    for (int c = 0; c < OUTC; ++c) s += __expf(v[c] - m);
    const float ls = __logf(s);
#pragma unroll
    for (int c = 0; c < OUTC; ++c)
        out[(size_t)node * OUTC + c] = v[c] - m - ls;
}

// ---------------------------------------------------------------------------
extern "C" void kernel_launch(void* const* d_in, const int* in_sizes, int n_in,
                              void* d_out, int out_size, void* d_ws, size_t ws_size,
                              hipStream_t stream) {
    const float* x  = (const float*)d_in[0];
    const float* W1 = (const float*)d_in[1];
    const float* b1 = (const float*)d_in[2];
    const float* W2 = (const float*)d_in[3];
    const float* b2 = (const float*)d_in[4];
    const int* edge = (const int*)d_in[5];

    const int N = in_sizes[0] / IN_CH;   // 100000
    const int E = in_sizes[5] / 2;       // 3200000
    const int* erow = edge;              // sources
    const int* ecol = edge + E;          // targets

    // workspace layout (floats): dis[N] | h1[N*16] | agg1[N*16] | h2[N*40]
    float* ws   = (float*)d_ws;
    float* dis  = ws;
    float* h1   = dis  + (size_t)N;
    float* agg1 = h1   + (size_t)N * HID;
    float* h2   = agg1 + (size_t)N * HID;
    float* out  = (float*)d_out;

    const int threads = 256;
    const int n_tiles = N / 16;                         // N is a multiple of 16
    const int gemm_blocks = (n_tiles + (threads / 32) - 1) / (threads / 32);

    gcn_init_kernel<<<(N * OUTC + threads - 1) / threads, threads, 0, stream>>>(dis, agg1, out, N);
    gcn_deg_kernel<<<(E + threads - 1) / threads, threads, 0, stream>>>(ecol, dis, E);
    gcn_dis_kernel<<<(N + threads - 1) / threads, threads, 0, stream>>>(dis, N);

    gcn_gemm1_kernel<<<gemm_blocks, threads, 0, stream>>>(x, W1, h1, n_tiles);
    gcn_scatter1_kernel<<<(E + threads - 1) / threads, threads, 0, stream>>>(erow, ecol, dis, h1, agg1, E);
    gcn_relu1_kernel<<<(N * HID + threads - 1) / threads, threads, 0, stream>>>(dis, h1, b1, agg1, N);

    gcn_gemm2_kernel<<<gemm_blocks, threads, 0, stream>>>(agg1, W2, h2, n_tiles);
    gcn_scatter2_kernel<<<(E + threads - 1) / threads, threads, 0, stream>>>(erow, ecol, dis, h2, out, E);
    gcn_finalize_kernel<<<(N + threads - 1) / threads, threads, 0, stream>>>(dis, h2, b2, out, N);
}